// AlphaNet_49589692399673
// MI455X (gfx1250) — compile-verified
//
#include <hip/hip_runtime.h>
#include <hip/hip_bf16.h>

// ---------------------------------------------------------------------------
// AlphaNet on MI455X (gfx1250), wave32 + V_WMMA_F32_16X16X4_F32.
// All BatchNorms are folded (exactly, including the two-level eps) into an
// effective W1/b1, reducing the net to: per-sample stats -> 50000x704x32 GEMM.
// ---------------------------------------------------------------------------

#define BTRUE  50000
#define HH     9
#define WW     30
#define SS     3
#define DD     10
#define NPAIR  36
#define NF     702
#define NFP    704      // K padded to multiple of 4
#define HID    30
#define NHP    32       // N padded to 2 tiles of 16
#define SPB    64       // samples per block
#define NBLK   ((BTRUE + SPB - 1) / SPB)   // 782
#define RS     708      // LDS feature row stride (even -> float2-aligned)
#define NGROUP 28
#define BN_EPS 1e-5f

typedef __attribute__((ext_vector_type(2))) float v2f;
typedef __attribute__((ext_vector_type(8))) float v8f;

// BN group of feature index j: group = conv*4 + {0 bn0, 1 max, 2 avg, 3 min}
__device__ __forceinline__ int group_of(int j) {
    int conv, base, K;
    if (j < 216)      { conv = 0; base = 0;   K = NPAIR; }
    else if (j < 432) { conv = 1; base = 216; K = NPAIR; }
    else {
        conv = 2 + (j - 432) / 54;
        base = 432 + (conv - 2) * 54;
        K = HH;
    }
    int off = j - base;
    int kind = (off < K * SS) ? 0 : 1 + (off - K * SS) / K;
    return conv * 4 + kind;
}

// Per-sample raw feature extraction. emit(j, group, value). Fully unrolled so
// every array index is a compile-time constant (all state stays in VGPRs).
template <typename EMIT>
__device__ __forceinline__ void compute_features(const float* __restrict__ gz, EMIT emit) {
    float m[HH][SS];
    float isd[HH][SS];
    const int zbase[5] = {432, 486, 540, 594, 648}; // sZ, decay, zscore, ret, mZ

    #pragma unroll
    for (int h = 0; h < HH; ++h) {
        float zr[WW];
        #pragma unroll
        for (int w = 0; w < WW; ++w) zr[w] = gz[h * WW + w];
        float mx[5], mn[5], sm[5];
        #pragma unroll
        for (int c = 0; c < 5; ++c) { mx[c] = -3.4e38f; mn[c] = 3.4e38f; sm[c] = 0.f; }
        #pragma unroll
        for (int s = 0; s < SS; ++s) {
            const int w0 = s * DD;
            float sum = 0.f, sq = 0.f, dec = 0.f;
            #pragma unroll
            for (int k = 0; k < DD; ++k) {
                float v = zr[w0 + k];
                sum += v; sq += v * v; dec += v * (float)(k + 1);
            }
            float mean  = sum * (1.f / DD);
            float var   = (sq - (float)DD * mean * mean) * (1.f / (DD - 1)); // ddof=1
            float sdev  = sqrtf(var);
            float inv   = 1.f / sdev;
            float decay = dec * (1.f / (0.5f * DD * (DD + 1)));
            float zsc   = mean * inv;
            float ret   = zr[w0 + DD - 1] / zr[w0] - 1.f;
            m[h][s] = mean; isd[h][s] = inv;
            float vals[5] = {sdev, decay, zsc, ret, mean};
            #pragma unroll
            for (int c = 0; c < 5; ++c) {
                emit(zbase[c] + h * SS + s, 8 + 4 * c, vals[c]);
                mx[c] = fmaxf(mx[c], vals[c]);
                mn[c] = fminf(mn[c], vals[c]);
                sm[c] += vals[c];
            }
        }
        #pragma unroll
        for (int c = 0; c < 5; ++c) {
            emit(zbase[c] + 27 + h, 8 + 4 * c + 1, mx[c]);
            emit(zbase[c] + 36 + h, 8 + 4 * c + 2, sm[c] * (1.f / SS));
            emit(zbase[c] + 45 + h, 8 + 4 * c + 3, mn[c]);
        }
    }
    // pair features (cov, corr) for all (x<y) row pairs, reference ordering
    int p = 0;
    #pragma unroll
    for (int x = 0; x < HH - 1; ++x) {
        float zx[WW];
        #pragma unroll
        for (int w = 0; w < WW; ++w) zx[w] = gz[x * WW + w];
        #pragma unroll
        for (int y = x + 1; y < HH; ++y) {
            float zy[WW];
            #pragma unroll
            for (int w = 0; w < WW; ++w) zy[w] = gz[y * WW + w];
            float cmx = -3.4e38f, cmn = 3.4e38f, csm = 0.f;
            float rmx = -3.4e38f, rmn = 3.4e38f, rsm = 0.f;
            #pragma unroll
            for (int s = 0; s < SS; ++s) {
                const int w0 = s * DD;
                float cr = 0.f;
                #pragma unroll
                for (int k = 0; k < DD; ++k) cr += zx[w0 + k] * zy[w0 + k];
                float cov  = (cr - (float)DD * m[x][s] * m[y][s]) * (1.f / (DD - 1));
                float corr = cov * isd[x][s] * isd[y][s];
                emit(p * SS + s,       0, cov);
                emit(216 + p * SS + s, 4, corr);
                cmx = fmaxf(cmx, cov);  cmn = fminf(cmn, cov);  csm += cov;
                rmx = fmaxf(rmx, corr); rmn = fminf(rmn, corr); rsm += corr;
            }
            emit(108 + p, 1, cmx); emit(144 + p, 2, csm * (1.f / SS)); emit(180 + p, 3, cmn);
            emit(324 + p, 5, rmx); emit(360 + p, 6, rsm * (1.f / SS)); emit(396 + p, 7, rmn);
            ++p;
        }
    }
}

// Kernel 1: per-block partial (sum, sum^2) for each of the 28 BN groups.
__global__ void __launch_bounds__(64) fs_kernel(const float* __restrict__ data,
                                                float* __restrict__ partials) {
    int tid = threadIdx.x;
    int b = blockIdx.x * SPB + tid;
    float s1[NGROUP], s2[NGROUP];
    #pragma unroll
    for (int g = 0; g < NGROUP; ++g) { s1[g] = 0.f; s2[g] = 0.f; }
    if (b < BTRUE) {
        const float* gz = data + (size_t)b * (HH * WW);
        compute_features(gz, [&](int j, int g, float v) { s1[g] += v; s2[g] += v * v; });
    }
    __shared__ float wsum[2][2 * NGROUP];
    int lane = tid & 31, wv = tid >> 5;
    #pragma unroll
    for (int i = 0; i < 2 * NGROUP; ++i) {
        float v = (i < NGROUP) ? s1[i] : s2[i - NGROUP];
        #pragma unroll
        for (int msk = 16; msk; msk >>= 1) v += __shfl_xor(v, msk, 32);
        if (lane == 0) wsum[wv][i] = v;
    }
    __syncthreads();
    if (tid < 2 * NGROUP)
        partials[(size_t)tid * NBLK + blockIdx.x] = wsum[0][tid] + wsum[1][tid];
}

// Kernel 2: reduce partials (fixed order, double precision) -> per-group
// final (scale, shift) with exact two-level BN eps handling.
__global__ void bn_kernel(const float* __restrict__ partials,
                          const float* __restrict__ gammap,
                          const float* __restrict__ betap,
                          float* __restrict__ stats) {
    __shared__ double smu[NGROUP], svar[NGROUP];
    int g = threadIdx.x;
    if (g < NGROUP) {
        double a1 = 0.0, a2 = 0.0;
        for (int blk = 0; blk < NBLK; ++blk) {
            a1 += (double)partials[(size_t)g * NBLK + blk];
            a2 += (double)partials[(size_t)(NGROUP + g) * NBLK + blk];
        }
        int conv = g >> 2, kind = g & 3;
        int K = (conv < 2) ? NPAIR : HH;
        double N = (double)BTRUE * K * ((kind == 0) ? SS : 1);
        double mu = a1 / N;
        double var = a2 / N - mu * mu;
        if (var < 0.0) var = 0.0;
        smu[g] = mu; svar[g] = var;
    }
    __syncthreads();
    if (g < NGROUP) {
        double gamma = (double)gammap[0], beta = (double)betap[0];
        int conv = g >> 2, kind = g & 3;
        double scale, shift;
        if (kind == 0) {
            scale = gamma / sqrt(svar[g] + (double)BN_EPS);
        } else {
            // pooled tensor is a*raw + c with a = gamma*rsqrt(var0+eps); its BN
            // variance is a^2*var_raw, so eps is NOT scale-invariant here.
            double a = gamma / sqrt(svar[conv * 4] + (double)BN_EPS);
            scale = gamma * a / sqrt(a * a * svar[g] + (double)BN_EPS);
        }
        shift = beta - scale * smu[g];
        stats[g] = (float)scale;
        stats[NGROUP + g] = (float)shift;
    }
}

// Kernel 3: fold BN into W1/b1. W1' stored K-pair-major matching the WMMA
// B-fragment striping: float2 at [pair=j/2][col] = {W1'[2p][col], W1'[2p+1][col]}.
__global__ void prep_kernel(const float* __restrict__ W1, const float* __restrict__ b1,
                            const float* __restrict__ stats,
                            float* __restrict__ w1e, float* __restrict__ b1e) {
    __shared__ float scl[NGROUP], shf[NGROUP];
    int tid = threadIdx.x;
    if (tid < NGROUP) { scl[tid] = stats[tid]; shf[tid] = stats[NGROUP + tid]; }
    __syncthreads();
    for (int idx = tid; idx < NFP * NHP; idx += blockDim.x) {
        int q = idx >> 6, rem = idx & 63;
        int col = rem >> 1, half = rem & 1;
        int j = 2 * q + half;
        float v = 0.f;
        if (j < NF && col < HID) v = W1[col * NF + j] * scl[group_of(j)];
        w1e[idx] = v;
    }
    for (int i = tid; i < NHP; i += blockDim.x) {
        if (i < HID) {
            float s = b1[i];
            for (int j = 0; j < NF; ++j) s += W1[i * NF + j] * shf[group_of(j)];
            b1e[i] = s;
        } else {
            b1e[i] = 0.f;
        }
    }
}

// Kernel 4: fused feature-recompute + WMMA GEMM (M=64/block, N=32, K=704)
// + relu + W2 reduction. 2 waves; each wave owns 2 M-tiles x 2 N-tiles.
__global__ void __launch_bounds__(64) gemm_kernel(const float* __restrict__ data,
                                                  const float* __restrict__ w1e,
                                                  const float* __restrict__ b1e,
                                                  const float* __restrict__ W2,
                                                  const float* __restrict__ b2p,
                                                  float* __restrict__ out) {
    __shared__ float feat[SPB * RS];   // 64 x 708 f32 feature tile
    __shared__ float hbuf[SPB * NHP];  // 64 x 32 f32 hidden tile
    int tid = threadIdx.x;
    int base = blockIdx.x * SPB;
    int b = base + tid;
    float* myrow = &feat[tid * RS];
    if (b < BTRUE) {
        myrow[NF] = 0.f; myrow[NF + 1] = 0.f;  // K padding
        compute_features(data + (size_t)b * (HH * WW),
                         [&](int j, int g, float v) { myrow[j] = v; });
    } else {
        for (int j = 0; j < NFP; ++j) myrow[j] = 0.f;
    }
    __syncthreads();

    int lane = tid & 31;
    int wv   = tid >> 5;       // wave 0: samples 0..31, wave 1: samples 32..63
    int lm   = lane & 15;
    int lh   = lane >> 4;
    const v2f* w1p = (const v2f*)w1e;  // [352 pairs][32 cols]
    v8f acc00 = {0.f,0.f,0.f,0.f,0.f,0.f,0.f,0.f};
    v8f acc01 = acc00, acc10 = acc00, acc11 = acc00;
    int rowA0 = (wv * 32 + lm) * RS;       // M-tile 0 of this wave
    int rowA1 = (wv * 32 + 16 + lm) * RS;  // M-tile 1 of this wave

    for (int k = 0; k < NFP; k += 4) {
        // A fragment (ISA 7.12.2: lane=M, vgpr pair = {K, K+1}, lane-half => K+2)
        v2f a0 = *(const v2f*)&feat[rowA0 + k + 2 * lh];
        v2f a1 = *(const v2f*)&feat[rowA1 + k + 2 * lh];
        // B fragment: pair row (k/2 + lane-half), cols lm / 16+lm
        const v2f* bp = w1p + (size_t)((k >> 1) + lh) * NHP;
        v2f b0 = bp[lm];
        v2f b1v = bp[16 + lm];
        acc00 = __builtin_amdgcn_wmma_f32_16x16x4_f32(false, a0, false, b0,  (short)0, acc00, false, false);
        acc01 = __builtin_amdgcn_wmma_f32_16x16x4_f32(false, a0, false, b1v, (short)0, acc01, false, false);
        acc10 = __builtin_amdgcn_wmma_f32_16x16x4_f32(false, a1, false, b0,  (short)0, acc10, false, false);
        acc11 = __builtin_amdgcn_wmma_f32_16x16x4_f32(false, a1, false, b1v, (short)0, acc11, false, false);
    }

    // scatter D fragments (vgpr r -> M=r / M=8+r per lane-half) to LDS
    #pragma unroll
    for (int r = 0; r < 8; ++r) {
        int m0 = wv * 32 + r + 8 * lh;
        hbuf[m0 * NHP + lm]      = acc00[r];
        hbuf[m0 * NHP + 16 + lm] = acc01[r];
        int m1 = m0 + 16;
        hbuf[m1 * NHP + lm]      = acc10[r];
        hbuf[m1 * NHP + 16 + lm] = acc11[r];
    }
    __syncthreads();

    if (b < BTRUE) {
        float o = b2p[0];
        #pragma unroll
        for (int i = 0; i < HID; ++i) {
            float h = hbuf[tid * NHP + i] + b1e[i];
            o += fmaxf(h, 0.f) * W2[i];
        }
        out[b] = o;
    }
}

extern "C" void kernel_launch(void* const* d_in, const int* in_sizes, int n_in,
                              void* d_out, int out_size, void* d_ws, size_t ws_size,
                              hipStream_t stream) {
    const float* data  = (const float*)d_in[0];
    const float* gamma = (const float*)d_in[1];
    const float* beta  = (const float*)d_in[2];
    const float* W1    = (const float*)d_in[3];
    const float* b1    = (const float*)d_in[4];
    const float* W2    = (const float*)d_in[5];
    const float* b2    = (const float*)d_in[6];
    float* out = (float*)d_out;
    float* ws  = (float*)d_ws;

    float* partials = ws;                        // 56 * NBLK floats
    float* stats    = partials + 2 * NGROUP * NBLK; // 56 floats
    float* w1e      = stats + 2 * NGROUP;        // NFP*NHP floats (8B aligned)
    float* b1e      = w1e + NFP * NHP;           // NHP floats

    fs_kernel<<<NBLK, SPB, 0, stream>>>(data, partials);
    bn_kernel<<<1, 32, 0, stream>>>(partials, gamma, beta, stats);
    prep_kernel<<<1, 256, 0, stream>>>(W1, b1, stats, w1e, b1e);
    gemm_kernel<<<NBLK, SPB, 0, stream>>>(data, w1e, b1e, W2, b2, out);
}